// DistanceAndAngle_7404523618474
// MI455X (gfx1250) — compile-verified
//
#include <hip/hip_runtime.h>
#include <math.h>

#define TPB   256
#define TILES 8
#define MAX_G 64

// gfx1250 async global->LDS path (ASYNCcnt), guarded so the file still
// compiles if the builtin signature differs on this toolchain.
#if defined(__HIP_DEVICE_COMPILE__) && defined(__gfx1250__) && \
    __has_builtin(__builtin_amdgcn_global_load_async_to_lds_b32) && \
    __has_builtin(__builtin_amdgcn_s_wait_asynccnt)
#define USE_ASYNC_LDS 1
typedef __attribute__((address_space(1))) int gas_int;  // global addr space
typedef __attribute__((address_space(3))) int las_int;  // LDS addr space
#endif

// ---------------------------------------------------------------------------
// Phase 1: one thread per edge. Computes pair vector + distance.
// Writes distance to d_out[0..E) and a packed float4 {vx,vy,vz,dist} table
// into d_ws so phase 2 does a single 128-bit gather per edge reference.
// Streams (edge_index, shift) use NT hints; table store stays RT so the
// 51.2 MB table remains resident in the 192 MB L2 for phase 2.
// ---------------------------------------------------------------------------
__global__ __launch_bounds__(TPB) void edge_kernel(
    const int* __restrict__ batch, const float* __restrict__ lattice,
    const float* __restrict__ pos, const int* __restrict__ eidx,
    const int* __restrict__ shift, float* __restrict__ dist,
    float4* __restrict__ table, int E, int G) {
  __shared__ float lat[MAX_G * 9];
  for (int k = threadIdx.x; k < G * 9; k += TPB) lat[k] = lattice[k];
  __syncthreads();

  int e = blockIdx.x * TPB + threadIdx.x;
  if (e >= E) return;

  int i = __builtin_nontemporal_load(eidx + e);
  int j = __builtin_nontemporal_load(eidx + E + e);
  float sx = (float)__builtin_nontemporal_load(shift + 3 * e + 0);
  float sy = (float)__builtin_nontemporal_load(shift + 3 * e + 1);
  float sz = (float)__builtin_nontemporal_load(shift + 3 * e + 2);

  int b = batch[i];
  const float* __restrict__ L = lat + 9 * b;
  // shift_vec[d] = sum_c s_c * L[c][d], pair = pos[j] + shift_vec - pos[i]
  float vx = pos[3 * j + 0] - pos[3 * i + 0] + sx * L[0] + sy * L[3] + sz * L[6];
  float vy = pos[3 * j + 1] - pos[3 * i + 1] + sx * L[1] + sy * L[4] + sz * L[7];
  float vz = pos[3 * j + 2] - pos[3 * i + 2] + sx * L[2] + sy * L[5] + sz * L[8];
  float d = sqrtf(vx * vx + vy * vy + vz * vz);

  __builtin_nontemporal_store(d, dist + e);   // not re-read: bypass cache
  if (table) table[e] = make_float4(vx, vy, vz, d);  // RT: keep in L2
}

// ---------------------------------------------------------------------------
// Phase 2: TILES tiles of TPB triplets per block. The triplet index stream
// (128 MB total) is double-buffered into LDS with async global->LDS copies;
// each lane reads back only the LDS word its own wave's async op wrote, so
// S_WAIT_ASYNCCNT alone (no barrier) synchronizes the pipeline. The float4
// table gathers hit L2; angles are stored non-temporally.
// ---------------------------------------------------------------------------
__global__ __launch_bounds__(TPB) void trip_kernel(
    const int* __restrict__ tei, const float4* __restrict__ table,
    float* __restrict__ ang, int T) {
#if USE_ASYNC_LDS
  __shared__ int s0[2][TPB];
  __shared__ int s1[2][TPB];
#endif
  const int tid = threadIdx.x;
  const long long base = (long long)blockIdx.x * (TPB * TILES);

#if USE_ASYNC_LDS
  {  // prefetch tile 0 into buffer 0 (clamped address: no divergent issue)
    long long idx = base + tid;
    long long q = idx < T ? idx : (long long)(T - 1);
    __builtin_amdgcn_global_load_async_to_lds_b32(
        (gas_int*)(tei + q), (las_int*)(&s0[0][tid]), 0, 0);
    __builtin_amdgcn_global_load_async_to_lds_b32(
        (gas_int*)(tei + T + q), (las_int*)(&s1[0][tid]), 0, 0);
  }
#endif

  for (int k = 0; k < TILES; ++k) {
    long long idx = base + (long long)k * TPB + tid;
    int e0, e1;
#if USE_ASYNC_LDS
    int buf = k & 1;
    if (k + 1 < TILES) {  // prefetch next tile, then wait for current one
      long long nidx = idx + TPB;
      long long q = nidx < T ? nidx : (long long)(T - 1);
      __builtin_amdgcn_global_load_async_to_lds_b32(
          (gas_int*)(tei + q), (las_int*)(&s0[buf ^ 1][tid]), 0, 0);
      __builtin_amdgcn_global_load_async_to_lds_b32(
          (gas_int*)(tei + T + q), (las_int*)(&s1[buf ^ 1][tid]), 0, 0);
      __builtin_amdgcn_s_wait_asynccnt(2);  // 2 in flight = next tile only
    } else {
      __builtin_amdgcn_s_wait_asynccnt(0);  // drain for last tile
    }
    asm volatile("" ::: "memory");  // compiler fence: LDS written by async
    e0 = s0[buf][tid];
    e1 = s1[buf][tid];
#else
    long long q = idx < T ? idx : (long long)(T - 1);
    e0 = __builtin_nontemporal_load(tei + q);
    e1 = __builtin_nontemporal_load(tei + T + q);
#endif
    float4 a = table[e0];  // single b128 gather, L2-resident
    float4 b = table[e1];
    float c = (a.x * b.x + a.y * b.y + a.z * b.z) / (a.w * b.w);
    c = c < -1.0f ? -1.0f : (c > 1.0f ? 1.0f : c);  // NaN propagates
    if (idx < T) __builtin_nontemporal_store(c, ang + idx);
  }
}

// ---------------------------------------------------------------------------
// Fallback (only if ws_size can't hold the E*16B table): recompute both
// edges per triplet.
// ---------------------------------------------------------------------------
__device__ __forceinline__ float4 edge_geom(int e, const int* __restrict__ batch,
                                            const float* __restrict__ lat,
                                            const float* __restrict__ pos,
                                            const int* __restrict__ eidx,
                                            const int* __restrict__ shift, int E) {
  int i = eidx[e], j = eidx[E + e];
  int b = batch[i];
  float sx = (float)shift[3 * e + 0];
  float sy = (float)shift[3 * e + 1];
  float sz = (float)shift[3 * e + 2];
  const float* __restrict__ L = lat + 9 * b;
  float vx = pos[3 * j + 0] - pos[3 * i + 0] + sx * L[0] + sy * L[3] + sz * L[6];
  float vy = pos[3 * j + 1] - pos[3 * i + 1] + sx * L[1] + sy * L[4] + sz * L[7];
  float vz = pos[3 * j + 2] - pos[3 * i + 2] + sx * L[2] + sy * L[5] + sz * L[8];
  float d = sqrtf(vx * vx + vy * vy + vz * vz);
  return make_float4(vx, vy, vz, d);
}

__global__ __launch_bounds__(TPB) void trip_fallback_kernel(
    const int* __restrict__ tei, const int* __restrict__ batch,
    const float* __restrict__ lattice, const float* __restrict__ pos,
    const int* __restrict__ eidx, const int* __restrict__ shift,
    float* __restrict__ ang, int E, int T, int G) {
  __shared__ float lat[MAX_G * 9];
  for (int k = threadIdx.x; k < G * 9; k += TPB) lat[k] = lattice[k];
  __syncthreads();
  long long t = (long long)blockIdx.x * TPB + threadIdx.x;
  if (t >= T) return;
  int e0 = __builtin_nontemporal_load(tei + t);
  int e1 = __builtin_nontemporal_load(tei + T + t);
  float4 a = edge_geom(e0, batch, lat, pos, eidx, shift, E);
  float4 b = edge_geom(e1, batch, lat, pos, eidx, shift, E);
  float c = (a.x * b.x + a.y * b.y + a.z * b.z) / (a.w * b.w);
  c = c < -1.0f ? -1.0f : (c > 1.0f ? 1.0f : c);
  ang[t] = c;
}

// ---------------------------------------------------------------------------
extern "C" void kernel_launch(void* const* d_in, const int* in_sizes, int n_in,
                              void* d_out, int out_size, void* d_ws, size_t ws_size,
                              hipStream_t stream) {
  const int*   batch   = (const int*)d_in[0];
  const float* lattice = (const float*)d_in[1];
  const float* pos     = (const float*)d_in[2];
  const int*   eidx    = (const int*)d_in[3];
  const int*   shift   = (const int*)d_in[4];
  const int*   tei     = (const int*)d_in[5];

  const int G = in_sizes[1] / 9;       // 64 graphs
  const int E = in_sizes[3] / 2;       // 3.2M edges
  const int T = in_sizes[5] / 2;       // 16M triplets

  float* dist = (float*)d_out;         // [E]
  float* ang  = dist + E;              // [T]

  const bool use_ws = ws_size >= (size_t)E * sizeof(float4);
  float4* table = use_ws ? (float4*)d_ws : nullptr;

  edge_kernel<<<(E + TPB - 1) / TPB, TPB, 0, stream>>>(
      batch, lattice, pos, eidx, shift, dist, table, E, G);

  if (use_ws) {
    const int per_block = TPB * TILES;
    trip_kernel<<<(T + per_block - 1) / per_block, TPB, 0, stream>>>(
        tei, table, ang, T);
  } else {
    trip_fallback_kernel<<<(T + TPB - 1) / TPB, TPB, 0, stream>>>(
        tei, batch, lattice, pos, eidx, shift, ang, E, T, G);
  }
}